// RNN_Model_26302379720942
// MI455X (gfx1250) — compile-verified
//
#include <hip/hip_runtime.h>
#include <math.h>

// B=64, S=2048, E=H=256, fp32 throughout (nonlinear 2048-step recurrence -> keep fp32,
// use V_WMMA_F32_16X16X4_F32).

typedef __attribute__((ext_vector_type(2))) float v2f;
typedef __attribute__((ext_vector_type(8))) float v8f;

#define LDSP 260   // padded LDS row stride (dwords): 16B-aligned rows, conflict-free frag reads

// ---------------------------------------------------------------------------
// Phase 1: xw[m,h] = sum_e x[m,e] * Ww[h,e] + Wb[h],  m in [0, B*S)
// Block = 32 M-rows x 256 N-cols, 512 threads = 16 waves.
// Wave w -> output columns [16w, 16w+16), two 16x16 M-tiles, K=256 in steps of 4.
// ---------------------------------------------------------------------------
__global__ __launch_bounds__(512)
void xw_gemm_kernel(const float* __restrict__ x, const float* __restrict__ Ww,
                    const float* __restrict__ Wb, float* __restrict__ xw)
{
    extern __shared__ float smem[];
    float* lds_w = smem;                 // Ww row-major [256][LDSP]
    float* lds_x = smem + 256 * LDSP;    // x strip     [32][LDSP]

    const int tid = threadIdx.x;
    const size_t m0 = (size_t)blockIdx.x * 32;

    // cooperative fill: all of Ww (256KB) -> LDS (L2-resident source)
    for (int i4 = tid; i4 < 256 * 256 / 4; i4 += 512) {
        int lin = i4 * 4;
        int h = lin >> 8, k = lin & 255;
        *(float4*)&lds_w[h * LDSP + k] = ((const float4*)Ww)[i4];
    }
    // x strip 32x256
    for (int i4 = tid; i4 < 32 * 256 / 4; i4 += 512) {
        int lin = i4 * 4;
        int m = lin >> 8, k = lin & 255;
        *(float4*)&lds_x[m * LDSP + k] = *(const float4*)&x[(m0 + m) * 256 + k];
    }
    __syncthreads();

    const int wave = tid >> 5;
    const int lane = tid & 31;
    const int n    = lane & 15;
    const int half = lane >> 4;          // 0: K+0/K+1 , 1: K+2/K+3 (per f32 A/B frag layout)
    const int hcol = wave * 16 + n;

    const float wb = Wb[hcol];
    v8f c0, c1;
#pragma unroll
    for (int r = 0; r < 8; ++r) { c0[r] = wb; c1[r] = wb; }

    const float* wrow  = &lds_w[hcol * LDSP];        // B[k][n] = Ww[hcol][k]
    const float* arow0 = &lds_x[n * LDSP];           // A tile 0: rows 0..15
    const float* arow1 = &lds_x[(16 + n) * LDSP];    // A tile 1: rows 16..31

#pragma unroll 4
    for (int k = 0; k < 256; k += 4) {
        const int kk = k + 2 * half;
        v2f b  = *(const v2f*)&wrow[kk];
        v2f a0 = *(const v2f*)&arow0[kk];
        v2f a1 = *(const v2f*)&arow1[kk];
        c0 = __builtin_amdgcn_wmma_f32_16x16x4_f32(false, a0, false, b, (short)0, c0, false, false);
        c1 = __builtin_amdgcn_wmma_f32_16x16x4_f32(false, a1, false, b, (short)0, c1, false, false);
    }

#pragma unroll
    for (int r = 0; r < 8; ++r) {
        const int mr = r + half * 8;                 // C layout: VGPR r -> M=r (+8 for hi half)
        xw[(m0 + mr) * 256 + hcol]      = c0[r];
        xw[(m0 + 16 + mr) * 256 + hcol] = c1[r];
    }
}

// ---------------------------------------------------------------------------
// Phase 2: sequential scan. Batch rows are independent -> 4 blocks x 16 rows.
// Uw lives in LDS for the whole scan; each wave additionally hoists ITS 16
// columns of Uw (64 v2f fragments = 128 VGPRs, wave32 allows up to 1024) into
// registers, so the per-step loop is: prefetch xw[t+1] (hides HBM latency
// under the WMMA chain) -> 64-long V_WMMA_F32_16X16X4_F32 accumulation chain
// (A from LDS, B from regs) -> fused O store + tanh into double-buffered h
// -> one barrier.  xw is staged in the O region of d_out, consumed in place.
// ---------------------------------------------------------------------------
__global__ __launch_bounds__(512)
void rnn_scan_kernel(const float* __restrict__ Uw, const float* __restrict__ Ub,
                     float* __restrict__ O /* in: xw, out: O */,
                     float* __restrict__ tfinal, int S)
{
    extern __shared__ float smem[];
    float* lds_u = smem;                 // Uw row-major [256][LDSP]
    float* hbuf  = smem + 256 * LDSP;    // double-buffered h: 2 x [16][LDSP]

    const int tid = threadIdx.x;
    const int b0  = blockIdx.x * 16;
    const int HS  = 16 * LDSP;

    for (int i4 = tid; i4 < 256 * 256 / 4; i4 += 512) {
        int lin = i4 * 4;
        int h = lin >> 8, k = lin & 255;
        *(float4*)&lds_u[h * LDSP + k] = ((const float4*)Uw)[i4];
    }
    // initial hidden input: h = tanh(xw[:,0,:] + 0)
    for (int i = tid; i < 16 * 256; i += 512) {
        int m = i >> 8, hh = i & 255;
        hbuf[m * LDSP + hh] = tanhf(O[(size_t)(b0 + m) * (size_t)S * 256 + hh]);
    }
    __syncthreads();

    const int wave = tid >> 5;
    const int lane = tid & 31;
    const int n    = lane & 15;
    const int half = lane >> 4;
    const int hcol = wave * 16 + n;
    const float ub = Ub[hcol];

    // Hoist the wave's B fragments (Uw columns [16w,16w+16), all K) to VGPRs.
    v2f bfrag[64];
    {
        const float* urow = &lds_u[hcol * LDSP];
#pragma unroll
        for (int i = 0; i < 64; ++i)
            bfrag[i] = *(const v2f*)&urow[4 * i + 2 * half];
    }

    // Per-lane 32-bit element offsets of O[b0+mr, 0, hcol] (byte range < 2^31).
    uint32_t goff[8];
#pragma unroll
    for (int r = 0; r < 8; ++r) {
        const int mr = r + half * 8;
        goff[r] = (uint32_t)(b0 + mr) * (uint32_t)(S * 256) + (uint32_t)hcol;
    }

    for (int t = 0; t < S; ++t) {
        const float* cur = hbuf + (t & 1) * HS;
        float*       nxt = hbuf + ((t + 1) & 1) * HS;

        // Prefetch xw[:, t+1, :] now; completes under the WMMA chain.
        const uint32_t tn = (uint32_t)((t + 1 < S) ? (t + 1) : t);
        float xnext[8];
#pragma unroll
        for (int r = 0; r < 8; ++r) xnext[r] = O[goff[r] + tn * 256u];

        v8f c;
#pragma unroll
        for (int r = 0; r < 8; ++r) c[r] = ub;

        const float* arow = &cur[n * LDSP];
#pragma unroll
        for (int k4 = 0; k4 < 64; ++k4) {
            v2f a = *(const v2f*)&arow[4 * k4 + 2 * half];
            c = __builtin_amdgcn_wmma_f32_16x16x4_f32(false, a, false, bfrag[k4],
                                                      (short)0, c, false, false);
        }

#pragma unroll
        for (int r = 0; r < 8; ++r) {
            const int mr = r + half * 8;
            const float tv = c[r];
            if (t + 1 < S) {
                nxt[mr * LDSP + hcol] = tanhf(xnext[r] + tv);   // next step's A
            } else {
                tfinal[(b0 + mr) * 256 + hcol] = tv;            // final state
            }
            O[goff[r] + (uint32_t)t * 256u] = tv;               // O[b, t, h]
        }
        __syncthreads();
    }
}

// ---------------------------------------------------------------------------
extern "C" void kernel_launch(void* const* d_in, const int* in_sizes, int n_in,
                              void* d_out, int out_size, void* d_ws, size_t ws_size,
                              hipStream_t stream)
{
    (void)in_sizes; (void)n_in; (void)out_size; (void)d_ws; (void)ws_size;

    const float* x  = (const float*)d_in[0];   // [64,2048,256]
    const float* Ww = (const float*)d_in[1];   // [256,256]
    const float* Wb = (const float*)d_in[2];   // [256]
    const float* Uw = (const float*)d_in[3];   // [256,256]
    const float* Ub = (const float*)d_in[4];   // [256]

    float* out    = (float*)d_out;
    float* tfinal = out;                       // [64,256]
    float* O      = out + 64 * 256;            // [64,2048,256]; also holds xw between phases

    const size_t lds1 = (size_t)(256 * LDSP + 32 * LDSP) * sizeof(float);      // ~300 KB
    xw_gemm_kernel<<<(64 * 2048) / 32, 512, lds1, stream>>>(x, Ww, Wb, O);

    const size_t lds2 = (size_t)(256 * LDSP + 2 * 16 * LDSP) * sizeof(float);  // ~300 KB
    rnn_scan_kernel<<<4, 512, lds2, stream>>>(Uw, Ub, O, tfinal, 2048);
}